// TinyGIN_18537078849981
// MI455X (gfx1250) — compile-verified
//
#include <hip/hip_runtime.h>

// ---------------------------------------------------------------------------
// TinyGIN forward for MI455X (gfx1250, wave32, WMMA).
//
// Roofline: edge gather/scatter = ~819MB random traffic + 204.8M f32 atomics
// dominates (HBM 23.3 TB/s). GEMMs are only 13.1 GFLOP total -> use fp32 WMMA
// (V_WMMA_F32_16X16X4_F32) for full precision at no perf cost. Weights are
// 64KB each -> L2-resident, streamed as B fragments. LDS used only for the
// D->A fragment re-layout between the two GEMMs (padded stride 132, bank-
// conflict free on 64 banks) and for block-local BN stat reduction.
// ---------------------------------------------------------------------------

#define N_NODES 100000
#define N_EDGES 1600000
#define DIM     128
#define LDST    132   // padded LDS row stride (floats): 132 % 64 = 4 -> no bank conflicts

typedef __attribute__((ext_vector_type(2))) float v2f;
typedef __attribute__((ext_vector_type(8))) float v8f;

// ---------------------------------------------------------------- zero fill
__global__ void zero_kernel(float* __restrict__ p, int n) {
  int i = blockIdx.x * blockDim.x + threadIdx.x;
  if (i < n) p[i] = 0.0f;
}

// ------------------------------------------------- edge scatter: agg[dst]+=x[src]
// One wave per edge: 32 lanes * float4 = 128 floats. Gather is a contiguous
// 512B row read; atomics go to 128 consecutive floats of the dst row.
__global__ void __launch_bounds__(256)
scatter_kernel(const float* __restrict__ x,
               const int* __restrict__ ei,      // (2,E): [0,E) = src, [E,2E) = dst
               float* __restrict__ agg) {
  int gid  = blockIdx.x * blockDim.x + threadIdx.x;
  int wave = gid >> 5;
  int lane = gid & 31;
  if (wave >= N_EDGES) return;
  int src = ei[wave];
  int dst = ei[N_EDGES + wave];
  const float4 v = ((const float4*)(x + (size_t)src * DIM))[lane];
  float* a = agg + (size_t)dst * DIM + lane * 4;
  atomicAdd(a + 0, v.x);
  atomicAdd(a + 1, v.y);
  atomicAdd(a + 2, v.z);
  atomicAdd(a + 3, v.w);
}

// ------------------------------------------------------- fused GIN MLP kernel
// 128 threads = 4 waves; each wave computes a 16-row x 128-col output tile:
//   hin = (1+eps)*x + agg
//   t   = relu(hin @ Wa + ba)      (GEMM1, fp32 WMMA 16x16x4, K-loop of 32)
//   out = t @ Wb + bb              (GEMM2 via LDS restage of t)
// plus per-column sum / sum-of-squares accumulation for BatchNorm.
__global__ void __launch_bounds__(128)
mlp_kernel(const float* __restrict__ xin,
           const float* __restrict__ agg,
           const float* __restrict__ eps_p,
           const float* __restrict__ Wa, const float* __restrict__ ba,
           const float* __restrict__ Wb, const float* __restrict__ bb,
           float* __restrict__ out,
           float* __restrict__ stats) {      // [0,128)=sum, [128,256)=sumsq
  __shared__ float stage[4 * 16 * LDST];     // per-wave 16x128 tile, padded
  __shared__ float s_sum[DIM];
  __shared__ float s_sq[DIM];

  const int tid  = threadIdx.x;
  const int wave = tid >> 5;
  const int lane = tid & 31;
  const int half = lane >> 4;   // 0: lanes 0-15, 1: lanes 16-31
  const int l16  = lane & 15;

  if (tid < DIM) { s_sum[tid] = 0.0f; s_sq[tid] = 0.0f; }
  __syncthreads();

  const int   rowBase = blockIdx.x * 64 + wave * 16;
  const float epse    = 1.0f + eps_p[0];
  float* __restrict__ myStage = stage + wave * 16 * LDST;

  // A-fragment row for this lane (clamped for the grid tail; stores are masked)
  int rowA = rowBase + l16;
  if (rowA >= N_NODES) rowA = N_NODES - 1;
  const float* __restrict__ xrow = xin + (size_t)rowA * DIM;
  const float* __restrict__ arow = agg + (size_t)rowA * DIM;

  // ---------------- GEMM1: acc = hin @ Wa ----------------
  v8f acc[8] = {};
  for (int kk = 0; kk < 32; ++kk) {
    const int k0 = kk * 4 + half * 2;      // even -> 8B aligned float2 loads
    const float2 xv = *(const float2*)(xrow + k0);
    const float2 av = *(const float2*)(arow + k0);
    v2f a;
    a.x = epse * xv.x + av.x;
    a.y = epse * xv.y + av.y;
#pragma unroll
    for (int t = 0; t < 8; ++t) {
      const int n = t * 16 + l16;
      v2f b;
      b.x = Wa[(size_t)k0 * DIM + n];
      b.y = Wa[(size_t)(k0 + 1) * DIM + n];
      acc[t] = __builtin_amdgcn_wmma_f32_16x16x4_f32(
          false, a, false, b, (short)0, acc[t], false, false);
    }
  }

  // bias + ReLU, restage D-layout -> row-major LDS tile (for GEMM2's A frags)
#pragma unroll
  for (int t = 0; t < 8; ++t) {
    const int   col  = t * 16 + l16;
    const float bias = ba[col];
#pragma unroll
    for (int r = 0; r < 8; ++r) {
      const int m = r + half * 8;                  // D layout: VGPR r <-> row m
      float v = acc[t][r] + bias;
      v = v > 0.0f ? v : 0.0f;
      myStage[m * LDST + col] = v;
    }
  }
  // wave-private tile: in-wave ds RAW ordering handled by s_wait_dscnt

  // ---------------- GEMM2: acc2 = t @ Wb ----------------
  v8f acc2[8] = {};
  for (int kk = 0; kk < 32; ++kk) {
    const int k0 = kk * 4 + half * 2;
    const v2f a = *(const v2f*)(myStage + l16 * LDST + k0);
#pragma unroll
    for (int t = 0; t < 8; ++t) {
      const int n = t * 16 + l16;
      v2f b;
      b.x = Wb[(size_t)k0 * DIM + n];
      b.y = Wb[(size_t)(k0 + 1) * DIM + n];
      acc2[t] = __builtin_amdgcn_wmma_f32_16x16x4_f32(
          false, a, false, b, (short)0, acc2[t], false, false);
    }
  }

  // bias, store, and accumulate BatchNorm statistics
#pragma unroll
  for (int t = 0; t < 8; ++t) {
    const int   col  = t * 16 + l16;
    const float bias = bb[col];
    float psum = 0.0f, psq = 0.0f;
#pragma unroll
    for (int r = 0; r < 8; ++r) {
      const int row = rowBase + r + half * 8;
      const float v = acc2[t][r] + bias;
      if (row < N_NODES) {
        out[(size_t)row * DIM + col] = v;
        psum += v;
        psq  += v * v;
      }
    }
    // lanes L and L+16 hold the same column -> combine halves (wave32 xor-16)
    psum += __shfl_xor(psum, 16, 32);
    psq  += __shfl_xor(psq, 16, 32);
    if (half == 0) {
      atomicAdd(&s_sum[col], psum);   // ds_add_f32
      atomicAdd(&s_sq[col],  psq);
    }
  }

  __syncthreads();
  if (tid < DIM) {                    // one global atomic per column per block
    atomicAdd(&stats[tid],        s_sum[tid]);
    atomicAdd(&stats[DIM + tid],  s_sq[tid]);
  }
}

// ---------------------------------------------- BatchNorm (batch stats) + ReLU
__global__ void __launch_bounds__(256)
bn_kernel(float* __restrict__ h,
          const float* __restrict__ stats,
          const float* __restrict__ gamma,
          const float* __restrict__ beta) {
  long long i = (long long)blockIdx.x * blockDim.x + threadIdx.x;
  if (i >= (long long)N_NODES * DIM) return;
  const int   j    = (int)(i & (DIM - 1));
  const float invN = 1.0f / (float)N_NODES;
  const float mean = stats[j] * invN;
  const float var  = stats[DIM + j] * invN - mean * mean;   // biased variance
  const float inv  = rsqrtf(var + 1e-5f);
  float v = (h[i] - mean) * inv * gamma[j] + beta[j];
  h[i] = v > 0.0f ? v : 0.0f;
}

// ---------------------------------------------------------------------------
extern "C" void kernel_launch(void* const* d_in, const int* in_sizes, int n_in,
                              void* d_out, int out_size, void* d_ws, size_t ws_size,
                              hipStream_t stream) {
  const float* x    = (const float*)d_in[0];
  const int*   ei   = (const int*)  d_in[1];
  const float* eps1 = (const float*)d_in[2];
  const float* eps2 = (const float*)d_in[3];
  const float* W1a  = (const float*)d_in[4];
  const float* b1a  = (const float*)d_in[5];
  const float* W1b  = (const float*)d_in[6];
  const float* b1b  = (const float*)d_in[7];
  const float* W2a  = (const float*)d_in[8];
  const float* b2a  = (const float*)d_in[9];
  const float* W2b  = (const float*)d_in[10];
  const float* b2b  = (const float*)d_in[11];
  const float* g1   = (const float*)d_in[12];
  const float* be1  = (const float*)d_in[13];
  const float* g2   = (const float*)d_in[14];
  const float* be2  = (const float*)d_in[15];

  float* agg   = (float*)d_ws;                       // N*D floats (51.2 MB)
  float* h1    = agg + (size_t)N_NODES * DIM;        // N*D floats (51.2 MB)
  float* stats = h1 + (size_t)N_NODES * DIM;         // 256 floats
  float* out   = (float*)d_out;

  const int ND        = N_NODES * DIM;
  const int zBlocks   = (ND + 255) / 256;
  const int sBlocks   = (N_EDGES + 7) / 8;           // 8 waves/block, 1 edge/wave
  const int mBlocks   = (N_NODES + 63) / 64;         // 4 waves/block, 16 rows/wave

  // ---- layer 1 ----
  zero_kernel<<<zBlocks, 256, 0, stream>>>(agg, ND);
  zero_kernel<<<1, 256, 0, stream>>>(stats, 2 * DIM);
  scatter_kernel<<<sBlocks, 256, 0, stream>>>(x, ei, agg);
  mlp_kernel<<<mBlocks, 128, 0, stream>>>(x, agg, eps1, W1a, b1a, W1b, b1b, h1, stats);
  bn_kernel<<<zBlocks, 256, 0, stream>>>(h1, stats, g1, be1);

  // ---- layer 2 ----
  zero_kernel<<<zBlocks, 256, 0, stream>>>(agg, ND);
  zero_kernel<<<1, 256, 0, stream>>>(stats, 2 * DIM);
  scatter_kernel<<<sBlocks, 256, 0, stream>>>(h1, ei, agg);
  mlp_kernel<<<mBlocks, 128, 0, stream>>>(h1, agg, eps2, W2a, b2a, W2b, b2b, out, stats);
  bn_kernel<<<zBlocks, 256, 0, stream>>>(out, stats, g2, be2);
}